// ExpandLossLayer_28243704938636
// MI455X (gfx1250) — compile-verified
//
#include <hip/hip_runtime.h>
#include <stdint.h>

#define B_DIM   4096
#define C_DIM   21
#define P_DIM   1681            // 41*41
#define NROWS   (B_DIM * C_DIM) // 86016
#define NSORT   2048            // pad to power of two for bitonic
#define THREADS 256

// ---------------------------------------------------------------------------
// Kernel 1: one workgroup per (b,c) row.
//   - async global->LDS stage of the 1681-float row (CDNA5 ASYNCcnt path)
//   - pad to 2048 with -inf
//   - bitonic sort descending in LDS
//   - weighted sums with geometric weights q^i (q=0.996 fg, q=0.999 bg)
//   - row max = sorted[0]
// ---------------------------------------------------------------------------
__global__ __launch_bounds__(THREADS) void row_sort_score_kernel(
    const float* __restrict__ g,
    float* __restrict__ fg_out,
    float* __restrict__ bg_out,
    float* __restrict__ mx_out)
{
    __shared__ float sm[NSORT];
    const int t   = threadIdx.x;
    const int row = blockIdx.x;
    const float* rowp = g + (size_t)row * P_DIM;

    // ---- Stage row into LDS with gfx1250 async loads (GVS mode: SGPR64 base
    //      + per-lane VGPR byte offset). Tracked by ASYNCcnt.
    {
        uint64_t base = (uint64_t)(uintptr_t)rowp;   // wave-uniform -> SGPR pair
        for (int i = t; i < P_DIM; i += THREADS) {
            uint32_t lds_addr = (uint32_t)(uintptr_t)(&sm[i]); // low 32 bits = LDS byte offset
            uint32_t goff     = (uint32_t)(i * 4);
            asm volatile("global_load_async_to_lds_b32 %0, %1, %2"
                         :: "v"(lds_addr), "v"(goff), "s"(base)
                         : "memory");
        }
    }
    // Pad tail with -inf (sorts to the back for descending order).
    for (int i = P_DIM + t; i < NSORT; i += THREADS)
        sm[i] = -__builtin_inff();

    // Drain async copies before anyone reads LDS, then workgroup barrier.
    asm volatile("s_wait_asynccnt 0" ::: "memory");
    __syncthreads();

    // ---- Bitonic sort, descending. 66 stages, 8 elements per thread.
    for (int k = 2; k <= NSORT; k <<= 1) {
        for (int j = k >> 1; j > 0; j >>= 1) {
            #pragma unroll
            for (int s = 0; s < NSORT / THREADS; ++s) {
                int i   = t + (s << 8);
                int ixj = i ^ j;
                if (ixj > i) {
                    float a = sm[i];
                    float b = sm[ixj];
                    bool desc = (i & k) == 0;       // final merge has (i&k)==0 -> descending
                    if ((a < b) == desc) { sm[i] = b; sm[ixj] = a; }
                }
            }
            __syncthreads();
        }
    }

    // ---- Weighted sums over sorted values: w[i] = q^i / sum(q^i, i<P).
    const float l2fg = log2f(0.996f);
    const float l2bg = log2f(0.999f);
    const float sum_fg = (1.0f - exp2f((float)P_DIM * l2fg)) / (1.0f - 0.996f);
    const float sum_bg = (1.0f - exp2f((float)P_DIM * l2bg)) / (1.0f - 0.999f);

    float facc = 0.0f, bacc = 0.0f;
    for (int i = t; i < P_DIM; i += THREADS) {
        float v  = sm[i];
        float fi = (float)i;
        facc = fmaf(v, exp2f(fi * l2fg), facc);
        bacc = fmaf(v, exp2f(fi * l2bg), bacc);
    }
    float mx = sm[0];          // sorted descending -> max at front
    __syncthreads();           // everyone done reading sm before reuse

    sm[t]           = facc;
    sm[THREADS + t] = bacc;
    __syncthreads();
    for (int off = THREADS >> 1; off > 0; off >>= 1) {
        if (t < off) {
            sm[t]           += sm[t + off];
            sm[THREADS + t] += sm[THREADS + t + off];
        }
        __syncthreads();
    }
    if (t == 0) {
        fg_out[row] = sm[0]       / sum_fg;
        bg_out[row] = sm[THREADS] / sum_bg;
        mx_out[row] = mx;
    }
}

// ---------------------------------------------------------------------------
// Kernel 2: combine per-row scores into the scalar loss. One block, fully
// deterministic tree reduction; writes d_out[0] unconditionally.
// ---------------------------------------------------------------------------
__global__ __launch_bounds__(1024) void loss_reduce_kernel(
    const int*   __restrict__ labels,
    const float* __restrict__ fg,
    const float* __restrict__ bg,
    const float* __restrict__ mx,
    float* __restrict__ out)
{
    __shared__ float red[1024];
    const int t = threadIdx.x;
    float acc = 0.0f;

    for (int b = t; b < B_DIM; b += 1024) {
        const int* lab = labels + b * C_DIM;
        const int rowbase = b * C_DIM;
        float loss = 0.0f;

        // background term (channel 0)
        if (lab[0] != 0) loss -= logf(bg[rowbase]);

        // foreground term (channels 1..20)
        int   nfg = 0; float sfg = 0.0f;
        for (int c = 1; c < C_DIM; ++c)
            if (lab[c] != 0) { ++nfg; sfg += logf(fg[rowbase + c]); }
        loss -= sfg / (float)nfg;   // labels[:,1]==1 guarantees nfg >= 1

        // absent term (all channels where label == 0)
        int   nab = 0; float sab = 0.0f;
        for (int c = 0; c < C_DIM; ++c)
            if (lab[c] == 0) { ++nab; sab += logf(mx[rowbase + c]); }
        loss -= sab / (float)nab;   // labels[:,2]==0 guarantees nab >= 1

        acc += loss;
    }

    red[t] = acc;
    __syncthreads();
    for (int off = 512; off > 0; off >>= 1) {
        if (t < off) red[t] += red[t + off];
        __syncthreads();
    }
    if (t == 0) out[0] = red[0] / (float)B_DIM;
}

// ---------------------------------------------------------------------------
extern "C" void kernel_launch(void* const* d_in, const int* in_sizes, int n_in,
                              void* d_out, int out_size, void* d_ws, size_t ws_size,
                              hipStream_t stream)
{
    const float* sm_mask = (const float*)d_in[0];   // (B, C, 41, 41) f32
    const int*   labels  = (const int*)d_in[1];     // (B, C) i32

    float* ws = (float*)d_ws;                       // needs 3*NROWS*4 B ~= 1.03 MB
    float* fg = ws;
    float* bg = ws + NROWS;
    float* mx = ws + 2 * NROWS;

    row_sort_score_kernel<<<NROWS, THREADS, 0, stream>>>(sm_mask, fg, bg, mx);
    loss_reduce_kernel<<<1, 1024, 0, stream>>>(labels, fg, bg, mx, (float*)d_out);
}